// S6DReal_71425306133129
// MI455X (gfx1250) — compile-verified
//
#include <hip/hip_runtime.h>
#include <hip/hip_bf16.h>
#include <math.h>

#define L_DIM 8192
#define C_DIM 256
#define S_DIM 64
#define K_DIM 256
#define NCH   32
#define CHL   (L_DIM / NCH)   // 256

typedef __attribute__((ext_vector_type(2))) float v2f;
typedef __attribute__((ext_vector_type(8))) float v8f;

// Numerically-stable softplus using only native v_exp_f32 / v_log_f32
// (base-2 hardware ops via amdgcn builtins):
// softplus(z) = max(z,0) + ln2 * log2(1 + exp2(-|z|*log2e))
__device__ __forceinline__ float softplus_fast(float z) {
    const float LOG2E = 1.44269504088896340736f;
    const float LN2   = 0.69314718055994530942f;
    float t = __builtin_amdgcn_exp2f(-fabsf(z) * LOG2E);   // v_exp_f32
    return fmaxf(z, 0.0f) + LN2 * __builtin_amdgcn_logf(1.0f + t); // v_log_f32
}

// ---------------------------------------------------------------------------
// GEMM: out[l,n] = epilogue( addc + sum_k X[l,k]*W[k,n] + bias[n] )
// One wave computes one 16x16 tile via V_WMMA_F32_16X16X4_F32, K looped by 4.
// ---------------------------------------------------------------------------
__global__ __launch_bounds__(128) void gemm_wmma_f32(
    const float* __restrict__ X,    // (L, K_DIM)
    const float* __restrict__ W,    // (K_DIM, N)
    const float* __restrict__ bias, // (N)
    float* __restrict__ out,        // (L, N)
    int N, float addc, int do_softplus)
{
    const int lane = threadIdx.x & 31;
    const int wave = threadIdx.x >> 5;
    const int tiles_n = N >> 4;
    const int tile = blockIdx.x * 4 + wave;
    const int tm = tile / tiles_n;
    const int tn = tile - tm * tiles_n;
    const int m0 = tm << 4;
    const int n0 = tn << 4;
    const int half = lane >> 4;      // which K-half of the 4-wide K block
    const int r    = lane & 15;      // row (A) / column (B,C,D) index

    v8f acc = {};
    const float* xrow = X + (size_t)(m0 + r) * K_DIM;

    for (int k0 = 0; k0 < K_DIM; k0 += 4) {
        const int k = k0 + half * 2;
        v2f a, b;
        // A 16x4: lane<16 -> K = k0+0,1 ; lane>=16 -> K = k0+2,3
        a.x = xrow[k + 0];
        a.y = xrow[k + 1];
        // B 4x16: V0 holds rows K=k0+{0,2}, V1 holds rows K=k0+{1,3}
        b.x = W[(size_t)(k + 0) * N + n0 + r];
        b.y = W[(size_t)(k + 1) * N + n0 + r];
        acc = __builtin_amdgcn_wmma_f32_16x16x4_f32(
            /*neg_a=*/false, a, /*neg_b=*/false, b,
            /*c_mod=*/(short)0, acc, /*reuse_a=*/false, /*reuse_b=*/false);
    }

    const float bv = bias[n0 + r] + addc;
#pragma unroll
    for (int v = 0; v < 8; ++v) {
        float z = acc[v] + bv;
        if (do_softplus) z = softplus_fast(z);
        out[(size_t)(m0 + v + 8 * half) * N + n0 + r] = z;
    }
}

// ---------------------------------------------------------------------------
// Pass 1: per (channel c, chunk j): local scan with h_in = 0.
// Stores chunk transfer (P = prod At, U = local final h) per (c,j,s).
// One wave (32 lanes) per (c,j); each lane owns s and s+32.
// ---------------------------------------------------------------------------
__global__ __launch_bounds__(32) void scan_partial(
    const float* __restrict__ X, const float* __restrict__ lognegA,
    const float* __restrict__ DT, const float* __restrict__ Bm,
    float* __restrict__ Pj, float* __restrict__ Uj)
{
    const int c = blockIdx.x & (C_DIM - 1);
    const int j = blockIdx.x >> 8;
    const int t = threadIdx.x;
    const int s1 = t, s2 = t + 32;

    const float a1 = -__expf(lognegA[c * S_DIM + s1]);
    const float a2 = -__expf(lognegA[c * S_DIM + s2]);
    const float ia1 = 1.0f / a1, ia2 = 1.0f / a2;

    float p1 = 1.0f, p2 = 1.0f, h1 = 0.0f, h2 = 0.0f;
    const int l0 = j * CHL;
    for (int i = 0; i < CHL; ++i) {
        const int l = l0 + i;
        // hide the stride-1KB latency of the broadcast x/dt streams
        if (l + 32 < L_DIM) {
            __builtin_prefetch(&X[(size_t)(l + 32) * C_DIM + c], 0, 0);
            __builtin_prefetch(&DT[(size_t)(l + 32) * C_DIM + c], 0, 0);
        }
        const float xv  = X[(size_t)l * C_DIM + c];
        const float dtv = DT[(size_t)l * C_DIM + c];
        const float e1 = __expf(a1 * dtv);
        const float e2 = __expf(a2 * dtv);
        const float u1 = (e1 - 1.0f) * Bm[(size_t)l * S_DIM + s1] * ia1 * xv;
        const float u2 = (e2 - 1.0f) * Bm[(size_t)l * S_DIM + s2] * ia2 * xv;
        h1 = e1 * h1 + u1;
        h2 = e2 * h2 + u2;
        p1 *= e1;
        p2 *= e2;
    }
    const size_t base = ((size_t)c * NCH + j) * S_DIM;
    Pj[base + s1] = p1; Pj[base + s2] = p2;
    Uj[base + s1] = h1; Uj[base + s2] = h2;
}

// ---------------------------------------------------------------------------
// Pass 2: serial combine across the 32 chunks per (c,s): tiny work.
// Hin[c,j,s] = state entering chunk j.
// ---------------------------------------------------------------------------
__global__ __launch_bounds__(256) void scan_carry(
    const float* __restrict__ Pj, const float* __restrict__ Uj,
    float* __restrict__ Hin)
{
    const int idx = blockIdx.x * blockDim.x + threadIdx.x; // c*64 + s
    if (idx >= C_DIM * S_DIM) return;
    const int c = idx >> 6;
    const int s = idx & 63;
    float carry = 0.0f;
    for (int j = 0; j < NCH; ++j) {
        const size_t base = ((size_t)c * NCH + j) * S_DIM + s;
        Hin[base] = carry;
        carry = Pj[base] * carry + Uj[base];
    }
}

// ---------------------------------------------------------------------------
// Pass 3: replay each chunk with its carry, contract with Cproj over s
// (wave32 shfl_xor tree), write y[l,c].
// ---------------------------------------------------------------------------
__global__ __launch_bounds__(32) void scan_final(
    const float* __restrict__ X, const float* __restrict__ lognegA,
    const float* __restrict__ DT, const float* __restrict__ Bm,
    const float* __restrict__ Cm, const float* __restrict__ Hin,
    float* __restrict__ Y)
{
    const int c = blockIdx.x & (C_DIM - 1);
    const int j = blockIdx.x >> 8;
    const int t = threadIdx.x;
    const int s1 = t, s2 = t + 32;

    const float a1 = -__expf(lognegA[c * S_DIM + s1]);
    const float a2 = -__expf(lognegA[c * S_DIM + s2]);
    const float ia1 = 1.0f / a1, ia2 = 1.0f / a2;

    const size_t base = ((size_t)c * NCH + j) * S_DIM;
    float h1 = Hin[base + s1];
    float h2 = Hin[base + s2];

    const int l0 = j * CHL;
    for (int i = 0; i < CHL; ++i) {
        const int l = l0 + i;
        if (l + 32 < L_DIM) {
            __builtin_prefetch(&X[(size_t)(l + 32) * C_DIM + c], 0, 0);
            __builtin_prefetch(&DT[(size_t)(l + 32) * C_DIM + c], 0, 0);
        }
        const float xv  = X[(size_t)l * C_DIM + c];
        const float dtv = DT[(size_t)l * C_DIM + c];
        const float e1 = __expf(a1 * dtv);
        const float e2 = __expf(a2 * dtv);
        const float u1 = (e1 - 1.0f) * Bm[(size_t)l * S_DIM + s1] * ia1 * xv;
        const float u2 = (e2 - 1.0f) * Bm[(size_t)l * S_DIM + s2] * ia2 * xv;
        h1 = e1 * h1 + u1;
        h2 = e2 * h2 + u2;
        float p = Cm[(size_t)l * S_DIM + s1] * h1 + Cm[(size_t)l * S_DIM + s2] * h2;
        // wave32 tree reduction over s
        for (int off = 16; off > 0; off >>= 1)
            p += __shfl_xor(p, off, 32);
        if (t == 0) Y[(size_t)l * C_DIM + c] = p;
    }
}

// ---------------------------------------------------------------------------
extern "C" void kernel_launch(void* const* d_in, const int* in_sizes, int n_in,
                              void* d_out, int out_size, void* d_ws, size_t ws_size,
                              hipStream_t stream) {
    const float* x       = (const float*)d_in[0];
    const float* lognegA = (const float*)d_in[1];
    const float* W_B     = (const float*)d_in[2];
    const float* b_B     = (const float*)d_in[3];
    const float* W_C     = (const float*)d_in[4];
    const float* b_C     = (const float*)d_in[5];
    const float* W_dt    = (const float*)d_in[6];
    const float* b_dt    = (const float*)d_in[7];
    float* y = (float*)d_out;

    float* ws   = (float*)d_ws;
    float* wsB  = ws;                                  // L*S
    float* wsC  = wsB  + (size_t)L_DIM * S_DIM;        // L*S
    float* wsdt = wsC  + (size_t)L_DIM * S_DIM;        // L*C
    float* Pj   = wsdt + (size_t)L_DIM * C_DIM;        // C*NCH*S
    float* Uj   = Pj   + (size_t)C_DIM * NCH * S_DIM;  // C*NCH*S
    float* Hin  = Uj   + (size_t)C_DIM * NCH * S_DIM;  // C*NCH*S

    // Projections via fp32 WMMA. tiles = (L/16)*(N/16); 4 waves (tiles)/block.
    {
        const int tiles64  = (L_DIM / 16) * (S_DIM / 16);   // 2048
        const int tiles256 = (L_DIM / 16) * (C_DIM / 16);   // 8192
        gemm_wmma_f32<<<tiles64 / 4, 128, 0, stream>>>(
            x, W_B, b_B, wsB, S_DIM, 1.0f, 0);
        gemm_wmma_f32<<<tiles64 / 4, 128, 0, stream>>>(
            x, W_C, b_C, wsC, S_DIM, 0.0f, 0);
        gemm_wmma_f32<<<tiles256 / 4, 128, 0, stream>>>(
            x, W_dt, b_dt, wsdt, C_DIM, 1.0f / 4096.0f, 1);
    }

    // 3-pass chunked scan.
    scan_partial<<<C_DIM * NCH, 32, 0, stream>>>(x, lognegA, wsdt, wsB, Pj, Uj);
    scan_carry<<<(C_DIM * S_DIM + 255) / 256, 256, 0, stream>>>(Pj, Uj, Hin);
    scan_final<<<C_DIM * NCH, 32, 0, stream>>>(x, lognegA, wsdt, wsB, wsC, Hin, y);
}